// TemporalGNN_82300163326608
// MI455X (gfx1250) — compile-verified
//
#include <hip/hip_runtime.h>
#include <hip/hip_bf16.h>
#include <math.h>
#include <stdint.h>

// ---------------- problem constants (match reference) ----------------
#define A_NODES 32
#define NBLK    128                // B in reference
#define NNODES  (A_NODES * NBLK)   // 4096
#define TSTEPS  5
#define KDIM    16
#define ADIM    8
#define NHEAD   4
#define DHEAD   64
#define DMODEL  256                // NHEAD*DHEAD
#define LDIM    64
#define FIN     24                 // KDIM + ADIM
#define TN      (TSTEPS * NNODES)  // 20480 rows for GAT gemms
#define BT      (NBLK * TSTEPS)    // 640 rows for QKV gemms
#define OUTW    400                // 64 + 64 + 256 + 16

typedef __attribute__((ext_vector_type(2))) float v2f;
typedef __attribute__((ext_vector_type(8))) float v8f;

// LDS byte offset of a __shared__ pointer (generic -> addrspace(3) -> int)
#define LDS_OFFSET(p) ((unsigned)(uintptr_t)(__attribute__((address_space(3))) const void*)(p))

// =====================================================================
// WMMA GEMM:  C[M x Nw] = A[M x Kdim] @ W[Kdim x Nw] (+ bias)
// One wave per 16x64 output strip (4 accumulators share one A fragment).
// grid = (M/16, Nw/64), block = 32 (one wave; EXEC all ones for WMMA).
//
// f32 WMMA fragment layouts (ISA 7.12.2):
//   A 16x4 : lanes 0-15 -> {K0,K1}, lanes 16-31 -> {K2,K3}, m = lane&15
//   B 4x16 : mirrored, n = lane&15
//   C/D    : row = vgpr + 8*(lane>=16), col = lane&15
// =====================================================================
__global__ __launch_bounds__(32)
void gemm_wmma(const float* __restrict__ A, const float* __restrict__ W,
               const float* __restrict__ bias, float* __restrict__ C,
               int M, int Kdim, int Nw) {
  const int mi   = blockIdx.x * 16;
  const int ni   = blockIdx.y * 64;
  const int l    = threadIdx.x;   // 0..31
  const int half = l >> 4;        // 0: lanes 0-15, 1: lanes 16-31
  const int lm   = l & 15;

  v8f acc[4];
#pragma unroll
  for (int j = 0; j < 4; ++j) acc[j] = (v8f){};

  // per-lane streaming pointers: plain adds in the K loop, no mul chains
  const float* ap = A + (size_t)(mi + lm) * Kdim + half * 2;
  const float* bp = W + (size_t)(half * 2) * Nw + ni + lm;

  for (int k0 = 0; k0 < Kdim; k0 += 4) {
    v2f a;
    a.x = ap[0];
    a.y = ap[1];
    ap += 4;
#pragma unroll
    for (int j = 0; j < 4; ++j) {
      v2f b;
      b.x = bp[j * 16];
      b.y = bp[j * 16 + Nw];
      acc[j] = __builtin_amdgcn_wmma_f32_16x16x4_f32(
          false, a, false, b, (short)0, acc[j], false, false);
    }
    bp += (size_t)4 * Nw;
  }

#pragma unroll
  for (int j = 0; j < 4; ++j) {
    const int n  = ni + j * 16 + lm;
    const float bv = bias ? bias[n] : 0.0f;
#pragma unroll
    for (int r = 0; r < 8; ++r) {
      const int m = mi + r + half * 8;
      C[(size_t)m * Nw + n] = acc[j][r] + bv;
    }
  }
}

// =====================================================================
// GAT attention + aggregation for one (t, block): dense 32x32 per head.
// Tile staged into LDS with the CDNA5 async-to-LDS path (ASYNCcnt).
// ego_only==0: out[(t*N + blk*32 + dst)*D + d] = relu(agg + bias)  (all dst)
// ego_only==1: out[(blk*T + t)*D + d]          = relu(agg + bias)  (dst==0)
// =====================================================================
__global__ __launch_bounds__(256)
void gat_block_attn(const float* __restrict__ h,      // (T*N, D)
                    const float* __restrict__ a_src,  // (H, DH)
                    const float* __restrict__ a_dst,  // (H, DH)
                    const float* __restrict__ bias,   // (D)
                    float* __restrict__ out,
                    int ego_only) {
  const int tb   = blockIdx.x;          // 0 .. T*NBLK-1
  const int t    = tb / NBLK;
  const int blk  = tb % NBLK;
  const int base = t * NNODES + blk * A_NODES;
  const int tid  = threadIdx.x;

  __shared__ float hs[A_NODES][DMODEL];   // 32 KB contiguous tile
  __shared__ float als[A_NODES][NHEAD];
  __shared__ float ald[A_NODES][NHEAD];

  // ---- async stage: 32x256 floats, contiguous in global, 16B per lane ----
  {
    const unsigned long long gbase =
        (unsigned long long)(uintptr_t)(h + (size_t)base * DMODEL);
    const unsigned lbase = LDS_OFFSET(&hs[0][0]);
#pragma unroll
    for (int it = 0; it < (A_NODES * DMODEL * 4) / (256 * 16); ++it) {  // 8 rounds
      const unsigned idx = (unsigned)(it * 256 + tid) * 16u;  // byte offset
      const unsigned laddr = lbase + idx;
      const unsigned long long gaddr = gbase + idx;
      asm volatile("global_load_async_to_lds_b128 %0, %1, off"
                   :: "v"(laddr), "v"(gaddr) : "memory");
    }
    asm volatile("s_wait_asynccnt 0x0" ::: "memory");
  }
  __syncthreads();

  // attention logits: 32 nodes x 4 heads; src half / dst half of the block
  if (tid < 128) {
    const int n = tid >> 2, hh = tid & 3;
    float s = 0.f;
    for (int d = 0; d < DHEAD; ++d) s += hs[n][hh * DHEAD + d] * a_src[hh * DHEAD + d];
    als[n][hh] = s;
  } else {
    const int u = tid - 128;
    const int n = u >> 2, hh = u & 3;
    float s = 0.f;
    for (int d = 0; d < DHEAD; ++d) s += hs[n][hh * DHEAD + d] * a_dst[hh * DHEAD + d];
    ald[n][hh] = s;
  }
  __syncthreads();

  // per-dst softmax over 32 sources + weighted aggregation
  // thread -> (dst = tid/8, dim chunk = tid%8 covering 32 dims, head = chunk/2)
  const int dst   = tid >> 3;
  const int chunk = tid & 7;
  const int hh    = chunk >> 1;
  const int d0    = chunk * 32;

  float e[A_NODES];
  float mx = -1e30f;
#pragma unroll
  for (int s = 0; s < A_NODES; ++s) {
    float x = als[s][hh] + ald[dst][hh];
    x = (x > 0.f) ? x : 0.2f * x;             // leaky_relu(., 0.2)
    e[s] = x;
    mx = fmaxf(mx, x);
  }
  float z = 0.f;
#pragma unroll
  for (int s = 0; s < A_NODES; ++s) { e[s] = expf(e[s] - mx); z += e[s]; }
  const float inv = 1.0f / (z + 1e-16f);

  float acc[32];
#pragma unroll
  for (int d = 0; d < 32; ++d) acc[d] = 0.f;
  for (int s = 0; s < A_NODES; ++s) {
    const float al = e[s] * inv;
#pragma unroll
    for (int d = 0; d < 32; ++d) acc[d] += al * hs[s][d0 + d];
  }

  if (!ego_only) {
#pragma unroll
    for (int d = 0; d < 32; ++d) {
      const float v = acc[d] + bias[d0 + d];
      out[(size_t)(base + dst) * DMODEL + d0 + d] = fmaxf(v, 0.f);
    }
  } else if (dst == 0) {
#pragma unroll
    for (int d = 0; d < 32; ++d) {
      const float v = acc[d] + bias[d0 + d];
      out[(size_t)(blk * TSTEPS + t) * DMODEL + d0 + d] = fmaxf(v, 0.f);
    }
  }
}

// =====================================================================
// Temporal attention (last query row only) + wo + output heads, per b.
// =====================================================================
__global__ __launch_bounds__(256)
void temporal_head(const float* __restrict__ q, const float* __restrict__ k,
                   const float* __restrict__ v,
                   const float* __restrict__ wo,  const float* __restrict__ bo,
                   const float* __restrict__ lmw, const float* __restrict__ lmb,
                   const float* __restrict__ lvw, const float* __restrict__ lvb,
                   const float* __restrict__ apw, const float* __restrict__ apb,
                   const float* __restrict__ bhw, const float* __restrict__ bhb,
                   float* __restrict__ out) {
  const int b   = blockIdx.x;
  const int tid = threadIdx.x;

  __shared__ float qL[DMODEL];
  __shared__ float kk[TSTEPS][DMODEL];
  __shared__ float vv[TSTEPS][DMODEL];
  __shared__ float sc[NHEAD][TSTEPS];
  __shared__ float ctx[DMODEL];
  __shared__ float feat[DMODEL];
  __shared__ float mu[LDIM];

  qL[tid] = q[(size_t)(b * TSTEPS + (TSTEPS - 1)) * DMODEL + tid];
  for (int i = tid; i < TSTEPS * DMODEL; i += 256) {
    const int s = i >> 8, d = i & 255;
    kk[s][d] = k[(size_t)(b * TSTEPS + s) * DMODEL + d];
    vv[s][d] = v[(size_t)(b * TSTEPS + s) * DMODEL + d];
  }
  __syncthreads();

  if (tid < NHEAD * TSTEPS) {                 // 20 score dots of length 64
    const int hh = tid / TSTEPS, s = tid % TSTEPS;
    float a = 0.f;
    for (int d = 0; d < DHEAD; ++d) a += qL[hh * DHEAD + d] * kk[s][hh * DHEAD + d];
    sc[hh][s] = a * 0.125f;                   // 1/sqrt(64); t=T-1 sees all s (causal)
  }
  __syncthreads();

  if (tid < NHEAD) {                          // softmax over 5
    float mx = -1e30f;
    for (int s = 0; s < TSTEPS; ++s) mx = fmaxf(mx, sc[tid][s]);
    float p[TSTEPS], z = 0.f;
    for (int s = 0; s < TSTEPS; ++s) { p[s] = expf(sc[tid][s] - mx); z += p[s]; }
    for (int s = 0; s < TSTEPS; ++s) sc[tid][s] = p[s] / z;
  }
  __syncthreads();

  {                                           // ctx (last timestep only)
    const int hh = tid >> 6;
    float a = 0.f;
    for (int s = 0; s < TSTEPS; ++s) a += sc[hh][s] * vv[s][tid];
    ctx[tid] = a;
  }
  __syncthreads();

  {                                           // feat = ctx @ wo + bo
    float a = bo[tid];
    for (int i = 0; i < DMODEL; ++i) a += ctx[i] * wo[(size_t)i * DMODEL + tid];
    feat[tid] = a;
  }
  __syncthreads();

  float* orow = out + (size_t)b * OUTW;
  if (tid < 64) {                             // mu
    float a = lmb[tid];
    for (int i = 0; i < DMODEL; ++i) a += feat[i] * lmw[(size_t)i * LDIM + tid];
    mu[tid] = a;
    orow[tid] = a;
  } else if (tid < 128) {                     // logvar
    const int j = tid - 64;
    float a = lvb[j];
    for (int i = 0; i < DMODEL; ++i) a += feat[i] * lvw[(size_t)i * LDIM + j];
    orow[64 + j] = a;
  } else if (tid < 144) {                     // belief
    const int j = tid - 128;
    float a = bhb[j];
    for (int i = 0; i < DMODEL; ++i) a += feat[i] * bhw[(size_t)i * KDIM + j];
    orow[384 + j] = a;
  }
  __syncthreads();

  {                                           // act = mu @ ap_w + ap_b
    float a = apb[tid];
    for (int j = 0; j < LDIM; ++j) a += mu[j] * apw[(size_t)j * DMODEL + tid];
    orow[128 + tid] = a;
  }
}

// =====================================================================
extern "C" void kernel_launch(void* const* d_in, const int* in_sizes, int n_in,
                              void* d_out, int out_size, void* d_ws, size_t ws_size,
                              hipStream_t stream) {
  const float* node_feats = (const float*)d_in[0];
  // d_in[1] = edge_index: fixed block-dense structure, not needed
  const float* g1w = (const float*)d_in[2];
  const float* g1as = (const float*)d_in[3];
  const float* g1ad = (const float*)d_in[4];
  const float* g1b = (const float*)d_in[5];
  const float* g2w = (const float*)d_in[6];
  const float* g2as = (const float*)d_in[7];
  const float* g2ad = (const float*)d_in[8];
  const float* g2b = (const float*)d_in[9];
  const float* wq = (const float*)d_in[10]; const float* bq = (const float*)d_in[11];
  const float* wk = (const float*)d_in[12]; const float* bk = (const float*)d_in[13];
  const float* wv = (const float*)d_in[14]; const float* bv = (const float*)d_in[15];
  const float* wo = (const float*)d_in[16]; const float* bo = (const float*)d_in[17];
  const float* lmw = (const float*)d_in[18]; const float* lmb = (const float*)d_in[19];
  const float* lvw = (const float*)d_in[20]; const float* lvb = (const float*)d_in[21];
  const float* apw = (const float*)d_in[22]; const float* apb = (const float*)d_in[23];
  const float* bhw = (const float*)d_in[24]; const float* bhb = (const float*)d_in[25];
  float* out = (float*)d_out;

  // workspace layout (floats): two 20 MB ping-pong buffers
  float* bufA = (float*)d_ws;                        // h1, later h2, later q/k/v
  float* bufB = bufA + (size_t)TN * DMODEL;          // x2, later seq
  float* seq  = bufB;                                // (B, T, D), overwrites freed x2
  float* qb   = bufA;                                // (B*T, D), overwrites freed h2
  float* kb   = bufA + (size_t)BT * DMODEL;
  float* vb   = bufA + (size_t)2 * BT * DMODEL;

  // 1) h1 = node_feats @ gat1_w       (20480 x 24 x 256)
  gemm_wmma<<<dim3(TN / 16, DMODEL / 64), 32, 0, stream>>>(
      node_feats, g1w, nullptr, bufA, TN, FIN, DMODEL);

  // 2) x2 = relu(agg(h1) + b1), all nodes
  gat_block_attn<<<TSTEPS * NBLK, 256, 0, stream>>>(
      bufA, g1as, g1ad, g1b, bufB, 0);

  // 3) h2 = x2 @ gat2_w               (20480 x 256 x 256)
  gemm_wmma<<<dim3(TN / 16, DMODEL / 64), 32, 0, stream>>>(
      bufB, g2w, nullptr, bufA, TN, DMODEL, DMODEL);

  // 4) seq[b,t,:] = relu(agg(h2) + b2) at ego nodes only
  gat_block_attn<<<TSTEPS * NBLK, 256, 0, stream>>>(
      bufA, g2as, g2ad, g2b, seq, 1);

  // 5) Q/K/V projections              (640 x 256 x 256 each, with bias)
  gemm_wmma<<<dim3(BT / 16, DMODEL / 64), 32, 0, stream>>>(seq, wq, bq, qb, BT, DMODEL, DMODEL);
  gemm_wmma<<<dim3(BT / 16, DMODEL / 64), 32, 0, stream>>>(seq, wk, bk, kb, BT, DMODEL, DMODEL);
  gemm_wmma<<<dim3(BT / 16, DMODEL / 64), 32, 0, stream>>>(seq, wv, bv, vb, BT, DMODEL, DMODEL);

  // 6) temporal attention (last row) + output heads
  temporal_head<<<NBLK, 256, 0, stream>>>(
      qb, kb, vb, wo, bo, lmw, lmb, lvw, lvb, apw, apb, bhw, bhb, out);
}